// CRF_12257836663177
// MI455X (gfx1250) — compile-verified
//
#include <hip/hip_runtime.h>
#include <hip/hip_bf16.h>
#include <stdint.h>

#define BB 64
#define SS 512
#define CC 256
#define CP 260   // padded row stride (floats) -> start bank = (4*tid + i) % 64, perfect b128

// Dynamic LDS layout (floats):
//   Tt   : [0, CC*CP)                 transposed transitions, Tt[j][i] = T[i][j]
//   alpha: [CC*CP, CC*CP + 2*CC)      double buffer
//   emit : [CC*CP + 2*CC, ... + 4*CC) double buffer (async-filled)
extern __shared__ float sm[];

__global__ __launch_bounds__(CC) void viterbi_fwd_kernel(
    const float* __restrict__ emissions,     // [B][S][C]
    const float* __restrict__ transitions,   // [C][C]
    unsigned char* __restrict__ bp,          // [B][S-1][C] backpointers (u8)
    int* __restrict__ last_tag)              // [B]
{
    const int b   = blockIdx.x;
    const int tid = threadIdx.x;

    float* Tt    = sm;
    float* alpha = sm + CC * CP;
    float* emitb = sm + CC * CP + 2 * CC;
    const uint32_t E0_bytes = (uint32_t)((CC * CP + 2 * CC) * 4);

    const float* embase = emissions + (size_t)b * SS * CC;

    // ---- Phase 0: stage transitions transposed into LDS (one-time, 256 KB) ----
    #pragma unroll 4
    for (int i = 0; i < CC; ++i) {
        Tt[tid * CP + i] = transitions[i * CC + tid];
    }
    // alpha_0 = emissions[:,0,:]
    alpha[tid] = embase[tid];

    // Prologue: async prefetch emissions[b,1,:] into emit buffer 1.
    // Each wave fills exactly its own lanes' slots -> per-wave asynccnt wait suffices.
    {
        uint32_t ldsoff = E0_bytes + (uint32_t)(((1 & 1) * CC + tid) * 4);
        uint32_t goff   = (uint32_t)((1 * CC + tid) * 4);
        asm volatile("global_load_async_to_lds_b32 %0, %1, %2"
                     :: "v"(ldsoff), "v"(goff), "s"(embase) : "memory");
    }
    __syncthreads();

    unsigned char* bpb = bp + (size_t)b * (SS - 1) * CC;

    // ---- Forward scan ----
    for (int t = 1; t < SS; ++t) {
        // Prefetch emissions for t+1 into the other emit buffer, then wait until
        // the load for step t landed (async loads complete in order).
        if (t + 1 < SS) {
            uint32_t ldsoff = E0_bytes + (uint32_t)((((t + 1) & 1) * CC + tid) * 4);
            uint32_t goff   = (uint32_t)(((t + 1) * CC + tid) * 4);
            asm volatile("global_load_async_to_lds_b32 %0, %1, %2"
                         :: "v"(ldsoff), "v"(goff), "s"(embase) : "memory");
            asm volatile("s_wait_asynccnt 1" ::: "memory");
        } else {
            asm volatile("s_wait_asynccnt 0" ::: "memory");
        }

        const float* aC = alpha + ((t - 1) & 1) * CC;
        float*       aN = alpha + (t & 1) * CC;

        const float4* trow = (const float4*)(Tt + tid * CP);  // 16B aligned (1040*tid)
        const float4* arow = (const float4*)aC;               // uniform broadcast reads

        // 4 independent argmax chains (i in [0,64),[64,128),[128,192),[192,256))
        // -> short dependence chains, VOPD-pairable; merged ascending with strict '>'
        // which reproduces jnp.argmax first-index tie-breaking exactly.
        float b0 = -INFINITY, b1 = -INFINITY, b2 = -INFINITY, b3 = -INFINITY;
        int   i0 = 0, i1 = 64, i2 = 128, i3 = 192;
        #pragma unroll 4
        for (int g = 0; g < 16; ++g) {
            float4 t0 = trow[g];      float4 a0 = arow[g];
            float4 t1 = trow[g + 16]; float4 a1 = arow[g + 16];
            float4 t2 = trow[g + 32]; float4 a2 = arow[g + 32];
            float4 t3 = trow[g + 48]; float4 a3 = arow[g + 48];
            float s;
            s = a0.x + t0.x; if (s > b0) { b0 = s; i0 = g * 4 + 0; }
            s = a1.x + t1.x; if (s > b1) { b1 = s; i1 = 64 + g * 4 + 0; }
            s = a2.x + t2.x; if (s > b2) { b2 = s; i2 = 128 + g * 4 + 0; }
            s = a3.x + t3.x; if (s > b3) { b3 = s; i3 = 192 + g * 4 + 0; }
            s = a0.y + t0.y; if (s > b0) { b0 = s; i0 = g * 4 + 1; }
            s = a1.y + t1.y; if (s > b1) { b1 = s; i1 = 64 + g * 4 + 1; }
            s = a2.y + t2.y; if (s > b2) { b2 = s; i2 = 128 + g * 4 + 1; }
            s = a3.y + t3.y; if (s > b3) { b3 = s; i3 = 192 + g * 4 + 1; }
            s = a0.z + t0.z; if (s > b0) { b0 = s; i0 = g * 4 + 2; }
            s = a1.z + t1.z; if (s > b1) { b1 = s; i1 = 64 + g * 4 + 2; }
            s = a2.z + t2.z; if (s > b2) { b2 = s; i2 = 128 + g * 4 + 2; }
            s = a3.z + t3.z; if (s > b3) { b3 = s; i3 = 192 + g * 4 + 2; }
            s = a0.w + t0.w; if (s > b0) { b0 = s; i0 = g * 4 + 3; }
            s = a1.w + t1.w; if (s > b1) { b1 = s; i1 = 64 + g * 4 + 3; }
            s = a2.w + t2.w; if (s > b2) { b2 = s; i2 = 128 + g * 4 + 3; }
            s = a3.w + t3.w; if (s > b3) { b3 = s; i3 = 192 + g * 4 + 3; }
        }
        // Merge chains (ascending index ranges, strict '>' keeps lowest index on ties)
        float best = b0; int besti = i0;
        if (b1 > best) { best = b1; besti = i1; }
        if (b2 > best) { best = b2; besti = i2; }
        if (b3 > best) { best = b3; besti = i3; }

        float emit = emitb[(t & 1) * CC + tid];        // own-wave async-filled slot
        bpb[(size_t)(t - 1) * CC + tid] = (unsigned char)besti;
        aN[tid] = best + emit;
        __syncthreads();   // single barrier: protects alpha double-buffer rotation
    }

    // ---- Final argmax over tags (tie-break: lowest index, matching jnp.argmax) ----
    float* vbuf = emitb;                 // reuse emit buffers as reduction scratch
    int*   ibuf = (int*)(emitb + CC);
    vbuf[tid] = alpha[((SS - 1) & 1) * CC + tid];
    ibuf[tid] = tid;
    __syncthreads();
    for (int off = CC / 2; off > 0; off >>= 1) {
        if (tid < off) {
            float vL = vbuf[tid], vR = vbuf[tid + off];
            int   iL = ibuf[tid], iR = ibuf[tid + off];
            if (vR > vL || (vR == vL && iR < iL)) { vbuf[tid] = vR; ibuf[tid] = iR; }
        }
        __syncthreads();
    }
    if (tid == 0) last_tag[b] = ibuf[0];
}

__global__ __launch_bounds__(BB) void viterbi_bwd_kernel(
    const unsigned char* __restrict__ bp,   // [B][S-1][C]
    const int* __restrict__ last_tag,       // [B]
    float* __restrict__ out)                // [B][S] (tags as float)
{
    const int PDIST = 16;                   // row-prefetch distance
    int b = threadIdx.x;
    if (b >= BB) return;
    int tag = last_tag[b];
    out[(size_t)b * SS + (SS - 1)] = (float)tag;
    const unsigned char* bpb = bp + (size_t)b * (SS - 1) * CC;

    // Prime prefetches: row addresses are known in advance (only the column
    // depends on the walked tag). global_prefetch_b8 covers the cacheline;
    // fetch both 128B halves of each 256B row.
    #pragma unroll
    for (int k = 0; k < PDIST; ++k) {
        int tp = SS - 2 - k;
        const unsigned char* p = bpb + (size_t)tp * CC;
        __builtin_prefetch(p, 0, 0);
        __builtin_prefetch(p + 128, 0, 0);
    }

    for (int t = SS - 2; t >= 0; --t) {
        int tp = t - PDIST;
        if (tp >= 0) {
            const unsigned char* p = bpb + (size_t)tp * CC;
            __builtin_prefetch(p, 0, 0);
            __builtin_prefetch(p + 128, 0, 0);
        }
        tag = (int)bpb[(size_t)t * CC + tag];
        out[(size_t)b * SS + t] = (float)tag;
    }
}

extern "C" void kernel_launch(void* const* d_in, const int* in_sizes, int n_in,
                              void* d_out, int out_size, void* d_ws, size_t ws_size,
                              hipStream_t stream) {
    const float* emissions   = (const float*)d_in[0];
    // d_in[1] = mask (all-ones; unused by reference math), d_in[2] = tags (unused)
    const float* transitions = (const float*)d_in[3];

    unsigned char* bp = (unsigned char*)d_ws;                       // 64*511*256 = 8,372,224 B
    int* last_tag     = (int*)((char*)d_ws + (size_t)BB * (SS - 1) * CC);

    const size_t ldsBytes = (size_t)(CC * CP + 4 * CC) * sizeof(float); // 270,336 B < 320 KB WGP LDS

    viterbi_fwd_kernel<<<dim3(BB), dim3(CC), ldsBytes, stream>>>(
        emissions, transitions, bp, last_tag);
    viterbi_bwd_kernel<<<dim3(1), dim3(BB), 0, stream>>>(
        bp, last_tag, (float*)d_out);
}